// GraphConvolution_20100446945335
// MI455X (gfx1250) — compile-verified
//
#include <hip/hip_runtime.h>
#include <hip/hip_bf16.h>

typedef __attribute__((ext_vector_type(2))) float v2f;
typedef __attribute__((ext_vector_type(8))) float v8f;

#define FEAT 256

// ---------------------------------------------------------------------------
// 1) Transpose weight [K][N] -> WT [N][K] so WMMA B-fragments (K-adjacent
//    pairs for a fixed N) are contiguous 64-bit loads.
// ---------------------------------------------------------------------------
__global__ void __launch_bounds__(256)
transpose_weight(const float* __restrict__ W, float* __restrict__ WT) {
    int idx = blockIdx.x * blockDim.x + threadIdx.x;   // 0 .. 65535
    int k = idx >> 8;
    int n = idx & 255;
    WT[(size_t)n * FEAT + k] = W[(size_t)k * FEAT + n];
}

// ---------------------------------------------------------------------------
// 2) Zero output (harness poisons d_out; scatter accumulates with atomics).
// ---------------------------------------------------------------------------
__global__ void __launch_bounds__(256)
zero_out(float4* __restrict__ out, int n4) {
    int i = blockIdx.x * blockDim.x + threadIdx.x;
    if (i < n4) out[i] = make_float4(0.f, 0.f, 0.f, 0.f);
}

// ---------------------------------------------------------------------------
// 3) support = input @ weight via V_WMMA_F32_16X16X4_F32 (wave32).
//    Each wave computes a 16x64 strip: 4 accumulators (v8f), K loop step 4.
//    A 16x4 fp32 layout (ISA 7.12.2): lanes 0-15 -> M=lane, K={k,k+1};
//    lanes 16-31 -> M=lane-16, K={k+2,k+3}.
//    B 4x16: VGPR0 = row K (lanes 0-15) / K+2 (lanes 16-31), VGPR1 = K+1/K+3,
//    i.e. per lane b = {WT[n][k+koff], WT[n][k+koff+1]} -> contiguous b64.
//    C/D 16x16 fp32: vgpr i, lanes 0-15 -> M=i, lanes 16-31 -> M=i+8.
// ---------------------------------------------------------------------------
__global__ void __launch_bounds__(256)
gemm_wmma_f32(const float* __restrict__ X, const float* __restrict__ WT,
              float* __restrict__ S) {
    const int lane = threadIdx.x & 31;
    const int wid  = threadIdx.x >> 5;
    const int gw   = blockIdx.x * 8 + wid;     // global wave id, 0..24999
    const int mbase = (gw >> 2) * 16;          // 6250 M tiles
    const int nbase = (gw & 3) * 64;           // 4 N groups of 64
    const int lmod  = lane & 15;
    const int koff  = (lane >> 4) * 2;         // 0 or 2

    const float* __restrict__ xrow = X  + (size_t)(mbase + lmod) * FEAT;
    const float* __restrict__ wt   = WT + (size_t)(nbase + lmod) * FEAT;

    v8f a0 = {}, a1 = {}, a2 = {}, a3 = {};

    for (int k = 0; k < FEAT; k += 4) {
        const int kk = k + koff;
        v2f av = *(const v2f*)(xrow + kk);
        v2f b0 = *(const v2f*)(wt + kk);
        v2f b1 = *(const v2f*)(wt + 16 * FEAT + kk);
        v2f b2 = *(const v2f*)(wt + 32 * FEAT + kk);
        v2f b3 = *(const v2f*)(wt + 48 * FEAT + kk);
        // 8 args: (neg_a, A, neg_b, B, c_mod, C, reuse_a, reuse_b)
        a0 = __builtin_amdgcn_wmma_f32_16x16x4_f32(false, av, false, b0, (short)0, a0, false, false);
        a1 = __builtin_amdgcn_wmma_f32_16x16x4_f32(false, av, false, b1, (short)0, a1, false, false);
        a2 = __builtin_amdgcn_wmma_f32_16x16x4_f32(false, av, false, b2, (short)0, a2, false, false);
        a3 = __builtin_amdgcn_wmma_f32_16x16x4_f32(false, av, false, b3, (short)0, a3, false, false);
    }

    const int rof = (lane >> 4) * 8;           // lanes 16-31 hold M = i+8
    #pragma unroll
    for (int i = 0; i < 8; ++i) {
        size_t base = (size_t)(mbase + rof + i) * FEAT + nbase + lmod;
        S[base +  0] = a0[i];
        S[base + 16] = a1[i];
        S[base + 32] = a2[i];
        S[base + 48] = a3[i];
    }
}

// ---------------------------------------------------------------------------
// 4) Edge scatter: one wave per edge. 32 lanes x 2 float4 loads cover the
//    256-float source row; 8 hardware fp32 atomics per lane into dst row.
//    support + out both fit in the 192MB L2 -> atomics resolve in L2.
// ---------------------------------------------------------------------------
__global__ void __launch_bounds__(256)
scatter_edges(const float* __restrict__ S, const long long* __restrict__ EI,
              const float* __restrict__ EW, float* __restrict__ O, int E) {
    const int gw   = (blockIdx.x * blockDim.x + threadIdx.x) >> 5;
    const int lane = threadIdx.x & 31;
    if (gw >= E) return;

    const long long s = EI[gw];
    const long long d = EI[(size_t)E + gw];
    const float     w = EW[gw];

    const float4* __restrict__ srow = (const float4*)(S + (size_t)s * FEAT);
    float* __restrict__ drow = O + (size_t)d * FEAT;

    float4 v0 = srow[lane];        // floats [4*lane .. 4*lane+3]
    float4 v1 = srow[lane + 32];   // floats [128+4*lane ..]

    const int b = lane * 4;
    unsafeAtomicAdd(drow + b + 0,        v0.x * w);
    unsafeAtomicAdd(drow + b + 1,        v0.y * w);
    unsafeAtomicAdd(drow + b + 2,        v0.z * w);
    unsafeAtomicAdd(drow + b + 3,        v0.w * w);
    unsafeAtomicAdd(drow + 128 + b + 0,  v1.x * w);
    unsafeAtomicAdd(drow + 128 + b + 1,  v1.y * w);
    unsafeAtomicAdd(drow + 128 + b + 2,  v1.z * w);
    unsafeAtomicAdd(drow + 128 + b + 3,  v1.w * w);
}

// ---------------------------------------------------------------------------
// 5) In-place vectorized ReLU epilogue.
// ---------------------------------------------------------------------------
__global__ void __launch_bounds__(256)
relu_inplace(float4* __restrict__ o, int n4) {
    int i = blockIdx.x * blockDim.x + threadIdx.x;
    if (i < n4) {
        float4 v = o[i];
        v.x = fmaxf(v.x, 0.f);
        v.y = fmaxf(v.y, 0.f);
        v.z = fmaxf(v.z, 0.f);
        v.w = fmaxf(v.w, 0.f);
        o[i] = v;
    }
}

extern "C" void kernel_launch(void* const* d_in, const int* in_sizes, int n_in,
                              void* d_out, int out_size, void* d_ws, size_t ws_size,
                              hipStream_t stream) {
    const float*     X  = (const float*)d_in[0];      // [100000, 256] f32
    const long long* EI = (const long long*)d_in[1];  // [2, E] i64
    const float*     EW = (const float*)d_in[2];      // [E] f32
    const float*     W  = (const float*)d_in[3];      // [256, 256] f32
    float*           out = (float*)d_out;             // [100000, 256] f32

    const int E = in_sizes[2];            // 3,200,000
    const int N = in_sizes[0] / FEAT;     // 100,000 (divisible by 16)

    // Workspace layout: WT (256*256 f32 = 256KB), then support (N*256 f32).
    float* WT = (float*)d_ws;
    float* S  = WT + FEAT * FEAT;

    transpose_weight<<<(FEAT * FEAT) / 256, 256, 0, stream>>>(W, WT);

    const int n4 = out_size / 4;
    zero_out<<<(n4 + 255) / 256, 256, 0, stream>>>((float4*)out, n4);

    const int waves  = (N / 16) * 4;      // 25,000 waves
    gemm_wmma_f32<<<waves / 8, 256, 0, stream>>>(X, WT, S);

    scatter_edges<<<(E * 32 + 255) / 256, 256, 0, stream>>>(S, EI, EW, out, E);

    relu_inplace<<<(n4 + 255) / 256, 256, 0, stream>>>((float4*)out, n4);
}